// GAT_84146999263862
// MI455X (gfx1250) — compile-verified
//
#include <hip/hip_runtime.h>

// ---------------- problem constants (match reference) ----------------
#define NN        50000          // nodes
#define NF        256            // input features
#define NH1       8              // heads layer 1
#define NC1       32             // channels per head layer 1  (NH1*NC1 == 256)
#define NHID      256            // NH1*NC1
#define NCLS      32             // classes (layer-2 channels, 1 head)
#define NE_IN     800000         // input edges
#define ETOT      (NE_IN + NN)   // + self loops
#define NEG_SLOPE 0.2f
#define MT        5              // 16-row C tiles stacked per wave (50000 = 625*80)

typedef float v2f __attribute__((ext_vector_type(2)));
typedef float v8f __attribute__((ext_vector_type(8)));

// ---------------------------------------------------------------------
// f32 WMMA GEMM:  C[M,N] = A[M,K] @ B[K,N]   (M % (16*MT)==0, N%16==0, K%4==0)
// One wave computes an 80x16 C macro-tile (MT stacked 16x16 tiles) with
// V_WMMA_F32_16X16X4_F32, reusing one B fragment across MT WMMAs per k-step.
// Fragment layouts per CDNA5 ISA 7.12.2:
//   A 16x4 : lane l -> row = l%16 ; vgpr r -> K = r + 2*(l/16)
//   B 4x16 : lane l -> col = l%16 ; vgpr r -> K = r + 2*(l/16)
//   C 16x16: lane l -> col = l%16 ; vgpr r -> row = r + 8*(l/16)
// ---------------------------------------------------------------------
__global__ void gat_gemm_wmma_f32(const float* __restrict__ A,
                                  const float* __restrict__ B,
                                  float* __restrict__ C,
                                  int M, int N, int K) {
  const int lane = threadIdx.x & 31;
  const int wave = threadIdx.x >> 5;
  const int wpb  = blockDim.x >> 5;
  const int tile = blockIdx.x * wpb + wave;
  const int ntiles  = N >> 4;
  const int mblocks = M / (16 * MT);
  if (tile >= mblocks * ntiles) return;

  const int tm   = (tile / ntiles) * (16 * MT);
  const int tn   = (tile % ntiles) << 4;
  const int half = lane >> 4;     // 0 or 1
  const int l16  = lane & 15;

  const float* arow = A + (long)(tm + l16) * K;
  const float* bcol = B + tn + l16;

  v8f acc[MT] = {};
#pragma unroll 2
  for (int k0 = 0; k0 < K; k0 += 4) {
    const int ka = k0 + 2 * half;
    v2f b;
    b.x = bcol[(long)ka * N];
    b.y = bcol[(long)(ka + 1) * N];
#pragma unroll
    for (int t = 0; t < MT; ++t) {
      const float* ap = arow + (long)t * 16 * K;
      v2f a;
      a.x = ap[ka];                 // contiguous pair -> global_load_b64
      a.y = ap[ka + 1];
      // (neg_a, A, neg_b, B, c_mod, C, reuse_a, reuse_b)
      acc[t] = __builtin_amdgcn_wmma_f32_16x16x4_f32(false, a, false, b,
                                                     (short)0, acc[t],
                                                     false, false);
    }
  }

#pragma unroll
  for (int t = 0; t < MT; ++t) {
    float* crow = C + (long)(tm + t * 16 + 8 * half) * N + tn + l16;
#pragma unroll
    for (int r = 0; r < 8; ++r)
      crow[(long)r * N] = acc[t][r];
  }
}

// ---------------------------------------------------------------------
__global__ void gat_fill(float* __restrict__ p, float v, int n) {
  int i = blockIdx.x * blockDim.x + threadIdx.x;
  if (i < n) p[i] = v;
}

// per-(node,head) attention coefficients: as[n,h] = z[n,h,:].a_src[h,:]
template <int H, int C>
__global__ void gat_node_alpha(const float* __restrict__ z,
                               const float* __restrict__ a_src,
                               const float* __restrict__ a_dst,
                               float* __restrict__ as_out,
                               float* __restrict__ ad_out) {
  int i = blockIdx.x * blockDim.x + threadIdx.x;   // i = node*H + h
  if (i >= NN * H) return;
  int h = i % H;
  const float* zp = z + (long)(i / H) * H * C + (long)h * C;
  float s = 0.f, d = 0.f;
#pragma unroll 4
  for (int c = 0; c < C; ++c) {
    float zv = zp[c];
    s += zv * a_src[h * C + c];
    d += zv * a_dst[h * C + c];
  }
  as_out[i] = s;
  ad_out[i] = d;
}

__device__ __forceinline__ void gat_edge(const int* __restrict__ ei, int e,
                                         int& s, int& d) {
  if (e < NE_IN) { s = ei[e]; d = ei[NE_IN + e]; }
  else           { s = d = e - NE_IN; }               // self loop
}

__device__ __forceinline__ float gat_leaky(float x) {
  return x > 0.f ? x : NEG_SLOPE * x;
}

// float atomic max via signed/unsigned integer atomics (correct for all signs)
__device__ __forceinline__ void gat_atomic_max_f32(float* addr, float v) {
  if (v >= 0.f) atomicMax((int*)addr, __float_as_int(v));
  else          atomicMin((unsigned int*)addr, __float_as_uint(v));
}

// pass 1: segment max of leaky_relu(as[src]+ad[dst]) per (dst, head)
template <int H>
__global__ void gat_edge_max(const int* __restrict__ ei,
                             const float* __restrict__ as,
                             const float* __restrict__ ad,
                             float* __restrict__ m) {
  int i = blockIdx.x * blockDim.x + threadIdx.x;
  if (i >= ETOT * H) return;
  int e = i / H, h = i - e * H;                      // H compile-time
  int s, d; gat_edge(ei, e, s, d);
  float logit = gat_leaky(as[s * H + h] + ad[d * H + h]);
  gat_atomic_max_f32(&m[d * H + h], logit);
}

// pass 2: segment sum of exp(logit - m[dst])
template <int H>
__global__ void gat_edge_expsum(const int* __restrict__ ei,
                                const float* __restrict__ as,
                                const float* __restrict__ ad,
                                const float* __restrict__ m,
                                float* __restrict__ ssum) {
  int i = blockIdx.x * blockDim.x + threadIdx.x;
  if (i >= ETOT * H) return;
  int e = i / H, h = i - e * H;
  int s, d; gat_edge(ei, e, s, d);
  float logit = gat_leaky(as[s * H + h] + ad[d * H + h]);
  atomicAdd(&ssum[d * H + h], __expf(logit - m[d * H + h]));
}

// pass 3: acc[dst,h,c0..c0+3] += z[src,h,c0..c0+3] * alpha(e,h)
// one thread = one float4 group of channels for one edge
template <int H, int C>
__global__ void gat_edge_agg(const int* __restrict__ ei,
                             const float* __restrict__ z,
                             const float* __restrict__ as,
                             const float* __restrict__ ad,
                             const float* __restrict__ m,
                             const float* __restrict__ ssum,
                             float* __restrict__ acc) {
  constexpr int G = (H * C) / 4;                     // float4 groups per edge
  int i = blockIdx.x * blockDim.x + threadIdx.x;     // i = e*G + g
  if (i >= ETOT * G) return;                         // max 54.4M, fits int32
  int e = i / G;                                     // compile-time shift
  int g = i - e * G;
  int h = (g * 4) / C;
  int s, d; gat_edge(ei, e, s, d);
  float logit = gat_leaky(as[s * H + h] + ad[d * H + h]);
  float alpha = __expf(logit - m[d * H + h]) / ssum[d * H + h];
  const float4 zv = *(const float4*)(z + (long)s * H * C + g * 4);
  float* ap = acc + (long)d * H * C + g * 4;
  atomicAdd(ap + 0, zv.x * alpha);
  atomicAdd(ap + 1, zv.y * alpha);
  atomicAdd(ap + 2, zv.z * alpha);
  atomicAdd(ap + 3, zv.w * alpha);
}

// h1 = elu(acc + b1)    (in place)
__global__ void gat_finalize1(float* __restrict__ h1,
                              const float* __restrict__ b1, int n) {
  int i = blockIdx.x * blockDim.x + threadIdx.x;
  if (i >= n) return;
  float v = h1[i] + b1[i % NHID];
  h1[i] = v > 0.f ? v : (__expf(v) - 1.f);
}

// out = acc2 + b2   (mean over 1 head == identity)
__global__ void gat_finalize2(const float* __restrict__ acc2,
                              const float* __restrict__ b2,
                              float* __restrict__ out, int n) {
  int i = blockIdx.x * blockDim.x + threadIdx.x;
  if (i >= n) return;
  out[i] = acc2[i] + b2[i % NCLS];
}

// ---------------------------------------------------------------------
extern "C" void kernel_launch(void* const* d_in, const int* in_sizes, int n_in,
                              void* d_out, int out_size, void* d_ws, size_t ws_size,
                              hipStream_t stream) {
  (void)in_sizes; (void)n_in; (void)out_size; (void)ws_size;

  const float* x     = (const float*)d_in[0];
  const int*   ei    = (const int*)  d_in[1];   // [2, NE_IN], int32
  const float* W1    = (const float*)d_in[2];
  const float* asrc1 = (const float*)d_in[3];
  const float* adst1 = (const float*)d_in[4];
  const float* b1    = (const float*)d_in[5];
  const float* W2    = (const float*)d_in[6];
  const float* asrc2 = (const float*)d_in[7];
  const float* adst2 = (const float*)d_in[8];
  const float* b2    = (const float*)d_in[9];
  float* out = (float*)d_out;

  // ---------------- workspace layout (floats) ----------------
  float* ws  = (float*)d_ws;
  float* z1  = ws;                       // 12,800,000   [NN, 256]
  float* h1  = ws + 12800000;            // 12,800,000   acc1 -> elu'd h1
  float* as1 = ws + 25600000;            //    400,000   [NN, 8]
  float* ad1 = as1 + 400000;             //    400,000
  float* m1  = ad1 + 400000;             //    400,000
  float* s1  = m1  + 400000;             //    400,000
  float* as2 = s1  + 400000;             //     50,000
  float* ad2 = as2 + 50000;
  float* m2  = ad2 + 50000;
  float* s2  = m2  + 50000;
  // layer-2 buffers reuse the (dead after layer-1 agg) z1 region:
  float* z2   = ws;                      //  1,600,000   [NN, 32]
  float* acc2 = ws + 1600000;            //  1,600,000

  const float NEG_BIG = -3.402823466e38f;
  auto blocks = [](long n, int t) { return (unsigned)((n + t - 1) / t); };

  // ---- init layer-1 accumulators ----
  gat_fill<<<blocks(12800000, 256), 256, 0, stream>>>(h1, 0.f, 12800000);
  gat_fill<<<blocks(400000, 256), 256, 0, stream>>>(m1, NEG_BIG, 400000);
  gat_fill<<<blocks(400000, 256), 256, 0, stream>>>(s1, 0.f, 400000);

  // ---- layer 1 ----
  {
    int M = NN, N = NHID, K = NF;                       // 50000 x 256 x 256
    int waves = (M / (16 * MT)) * (N / 16);             // 625 * 16
    gat_gemm_wmma_f32<<<blocks(waves, 4), 128, 0, stream>>>(x, W1, z1, M, N, K);
  }
  gat_node_alpha<NH1, NC1><<<blocks((long)NN * NH1, 256), 256, 0, stream>>>(
      z1, asrc1, adst1, as1, ad1);
  gat_edge_max<NH1><<<blocks((long)ETOT * NH1, 256), 256, 0, stream>>>(
      ei, as1, ad1, m1);
  gat_edge_expsum<NH1><<<blocks((long)ETOT * NH1, 256), 256, 0, stream>>>(
      ei, as1, ad1, m1, s1);
  gat_edge_agg<NH1, NC1><<<blocks((long)ETOT * (NHID / 4), 256), 256, 0, stream>>>(
      ei, z1, as1, ad1, m1, s1, h1);
  gat_finalize1<<<blocks((long)NN * NHID, 256), 256, 0, stream>>>(h1, b1, NN * NHID);

  // ---- layer 2 ----
  {
    int M = NN, N = NCLS, K = NHID;                     // 50000 x 32 x 256
    int waves = (M / (16 * MT)) * (N / 16);             // 625 * 2
    gat_gemm_wmma_f32<<<blocks(waves, 4), 128, 0, stream>>>(h1, W2, z2, M, N, K);
  }
  gat_fill<<<blocks(1600000, 256), 256, 0, stream>>>(acc2, 0.f, 1600000);
  gat_fill<<<blocks(50000, 256), 256, 0, stream>>>(m2, NEG_BIG, 50000);
  gat_fill<<<blocks(50000, 256), 256, 0, stream>>>(s2, 0.f, 50000);

  gat_node_alpha<1, NCLS><<<blocks(NN, 256), 256, 0, stream>>>(
      z2, asrc2, adst2, as2, ad2);
  gat_edge_max<1><<<blocks(ETOT, 256), 256, 0, stream>>>(ei, as2, ad2, m2);
  gat_edge_expsum<1><<<blocks(ETOT, 256), 256, 0, stream>>>(ei, as2, ad2, m2, s2);
  gat_edge_agg<1, NCLS><<<blocks((long)ETOT * (NCLS / 4), 256), 256, 0, stream>>>(
      ei, z2, as2, ad2, m2, s2, acc2);
  gat_finalize2<<<blocks((long)NN * NCLS, 256), 256, 0, stream>>>(
      acc2, b2, out, NN * NCLS);
}